// HMM_9380208575285
// MI455X (gfx1250) — compile-verified
//
#include <hip/hip_runtime.h>
#include <hip/hip_bf16.h>
#include <stdint.h>

#define NS 512      // states
#define NE 2048     // emissions
#define NB 64       // batch
#define NT 512      // time steps

// ---------------------------------------------------------------------------
// CDNA5 helpers: async global->LDS copy + ASYNCcnt wait (inline asm; no
// builtin covers GLOBAL_LOAD_ASYNC_TO_LDS on this toolchain).
// ---------------------------------------------------------------------------
template <int N>
__device__ __forceinline__ void wait_asynccnt() {
  asm volatile("s_wait_asynccnt %0" :: "i"(N) : "memory");
}

// Copy one 64-row x 32-col (8 KB) wave-private slice of the transition matrix
// into LDS. Global rows are 2048 B apart (full matrix row), LDS rows are 128 B
// (packed slice). Each lane moves 16 B per instruction; 16 instructions total.
__device__ __forceinline__ void async_copy_panel(const float* gsrc_row0,
                                                 float* ldst, int lane) {
  const int r  = lane >> 3;          // 0..3  : row within a 4-row quad
  const int cb = (lane & 7) << 4;    // 0..112: byte column within 128B row
  uint64_t g = (uint64_t)(uintptr_t)gsrc_row0 + (uint64_t)r * (NS * 4) + cb;
  uint32_t l = (uint32_t)(uintptr_t)ldst + (uint32_t)(r * 128 + cb);
#pragma unroll
  for (int q = 0; q < 16; ++q) {     // 16 quads * 4 rows = 64 rows
    asm volatile("global_load_async_to_lds_b128 %0, %1, off"
                 :: "v"(l), "v"(g) : "memory");
    g += (uint64_t)4 * NS * 4;       // advance 4 global rows
    l += 4 * 128;                    // advance 4 LDS rows
  }
}

// ---------------------------------------------------------------------------
// Forward pass. One workgroup (512 thr = 16 wave32) per batch. Thread tid
// owns output state j = tid. Wave w owns transition columns [32w, 32w+32);
// it streams 64-row panels of those columns through a wave-private LDS
// double buffer filled by async-to-LDS loads (per-wave ASYNCcnt sync only --
// zero intra-step workgroup barriers). The hot loop is argmax-free:
// add + max (2 VALU/elem); backpointers are recovered later from vhist.
// vhist[t][b][j] = score of best path ending in state j at time t.
// ---------------------------------------------------------------------------
__launch_bounds__(512, 1)
__global__ void viterbi_forward(const int*   __restrict__ obs,
                                const float* __restrict__ start,
                                const float* __restrict__ trans,
                                const float* __restrict__ emT,
                                float*       __restrict__ vhist) {
  extern __shared__ float lds[];
  // LDS layout:
  //   [0, 16*2*2048)          : per-wave double-buffered panel slices (256 KB)
  //   [65536, 65536 + 1024)   : double-buffered v vector (2 x 512 floats)
  const int b    = blockIdx.x;
  const int tid  = threadIdx.x;
  const int w    = tid >> 5;
  const int lane = tid & 31;
  const int j    = tid;

  float* slice0 = lds + w * 2 * 2048;        // 64 x 32 floats
  float* slice1 = slice0 + 2048;
  float* vbuf0  = lds + 16 * 2 * 2048;
  float* vbuf1  = vbuf0 + NS;

  // t = 0: v0 = start + emissions[:, obs[b,0]]
  {
    int o = obs[b * NT];
    float v0 = start[j] + emT[o * NS + j];
    vbuf0[j] = v0;
    vhist[(size_t)b * NS + j] = v0;
  }
  __syncthreads();

  const float* tw = trans + (w << 5);        // this wave's 32-column stripe
  async_copy_panel(tw, slice0, lane);        // prefetch panel 0 of t=1

  for (int t = 1; t < NT; ++t) {
    float* vp = (t & 1) ? vbuf0 : vbuf1;     // read  (t=1 reads vbuf0)
    float* vn = (t & 1) ? vbuf1 : vbuf0;     // write
    float m = -INFINITY;

#pragma unroll 1
    for (int p = 0; p < 8; ++p) {            // 8 panels x 64 rows = 512 i
      float* cur = (p & 1) ? slice1 : slice0;
      float* nxt = (p & 1) ? slice0 : slice1;
      // Prefetch panel (p+1) mod 8: for p==7 this is panel 0 of the NEXT
      // time step (addresses are t-invariant), hiding latency across the
      // step barrier. In-order ASYNCcnt => waiting <=16 retires panel p.
      async_copy_panel(tw + (size_t)(((p + 1) & 7) * 64) * NS, nxt, lane);
      wait_asynccnt<16>();

      const float4* vpp = (const float4*)(vp + p * 64);
#pragma unroll
      for (int i4 = 0; i4 < 16; ++i4) {
        float4 vv = vpp[i4];                 // ds_load_b128, lane-broadcast
        int base = i4 * 128 + lane;          // 4 rows x 32 floats
        m = fmaxf(m, vv.x + cur[base]);
        m = fmaxf(m, vv.y + cur[base + 32]);
        m = fmaxf(m, vv.z + cur[base + 64]);
        m = fmaxf(m, vv.w + cur[base + 96]);
      }
    }

    int o = obs[b * NT + t];                 // uniform -> scalar load
    float vnew = m + emT[o * NS + j];        // coalesced gather row
    vn[j] = vnew;
    vhist[(size_t)t * (NB * NS) + (size_t)b * NS + j] = vnew;
    __syncthreads();                         // publish vn for next step
  }
}

// ---------------------------------------------------------------------------
// Transpose emissions [S][E] -> emT [E][S] and transitions [S][S] -> trT
// (trT[s*S+i] = trans[i*S+s]) so later reads are unit-stride.
// ---------------------------------------------------------------------------
__global__ void prep_transpose(const float* __restrict__ emis,
                               const float* __restrict__ trans,
                               float* __restrict__ emT,
                               float* __restrict__ trT) {
  int idx = blockIdx.x * blockDim.x + threadIdx.x;   // over E*S
  int e = idx / NS;
  int s = idx - e * NS;
  emT[idx] = emis[s * NE + e];                        // coalesced store
  if (idx < NS * NS) {
    int i  = idx & (NS - 1);
    int ss = idx >> 9;
    trT[idx] = trans[i * NS + ss];
  }
}

// ---------------------------------------------------------------------------
// Backtrace. One wave32 per batch. Recompute the argmax only along the
// surviving path using the stored v history (exact: fp32 max/add round the
// same way the forward pass and the reference do; ties resolve to the
// first index like jnp.argmax).
// ---------------------------------------------------------------------------
__device__ __forceinline__ int wave_argmax(float best, int bi) {
#pragma unroll
  for (int off = 16; off > 0; off >>= 1) {
    float ob = __shfl_down(best, off);
    int   oi = __shfl_down(bi, off);
    if (ob > best || (ob == best && oi < bi)) { best = ob; bi = oi; }
  }
  return __shfl(bi, 0);
}

__global__ void viterbi_backtrace(const float* __restrict__ vhist,
                                  const float* __restrict__ trT,
                                  float*       __restrict__ out) {
  const int b    = blockIdx.x;
  const int lane = threadIdx.x;

  // state at T-1 = argmax_j vhist[T-1][b][j]
  const float* vl = vhist + (size_t)(NT - 1) * (NB * NS) + (size_t)b * NS;
  float best = -INFINITY; int bi = 0;
#pragma unroll
  for (int k = 0; k < NS / 32; ++k) {
    int i = lane + 32 * k;
    float x = vl[i];
    if (x > best) { best = x; bi = i; }      // ascending i keeps first max
  }
  int s = wave_argmax(best, bi);
  if (lane == 0) out[b * NT + (NT - 1)] = (float)s;

  for (int t = NT - 1; t >= 1; --t) {
    const float* vp = vhist + (size_t)(t - 1) * (NB * NS) + (size_t)b * NS;
    const float* tc = trT + (size_t)s * NS;  // trans[i][s], unit-stride in i
    float bv = -INFINITY; int ii = 0;
#pragma unroll
    for (int k = 0; k < NS / 32; ++k) {
      int i = lane + 32 * k;
      float x = vp[i] + tc[i];
      if (x > bv) { bv = x; ii = i; }
    }
    s = wave_argmax(bv, ii);
    if (lane == 0) out[b * NT + (t - 1)] = (float)s;
  }
}

// ---------------------------------------------------------------------------
extern "C" void kernel_launch(void* const* d_in, const int* in_sizes, int n_in,
                              void* d_out, int out_size, void* d_ws,
                              size_t ws_size, hipStream_t stream) {
  (void)in_sizes; (void)n_in; (void)out_size; (void)ws_size;
  const int*   obs   = (const int*)  d_in[0];   // [B, T]
  const float* start = (const float*)d_in[1];   // [S]
  const float* trans = (const float*)d_in[2];   // [S, S]
  const float* emis  = (const float*)d_in[3];   // [S, E]

  float* ws    = (float*)d_ws;
  float* emT   = ws;                                  // E*S   floats (4 MB)
  float* trT   = emT + (size_t)NE * NS;               // S*S   floats (1 MB)
  float* vhist = trT + (size_t)NS * NS;               // T*B*S floats (67 MB)

  prep_transpose<<<(NS * NE) / 256, 256, 0, stream>>>(emis, trans, emT, trT);

  const size_t shmem = (16 * 2 * 2048 + 2 * NS) * sizeof(float); // 266,240 B
  (void)hipFuncSetAttribute((const void*)viterbi_forward,
                            hipFuncAttributeMaxDynamicSharedMemorySize,
                            (int)shmem);
  viterbi_forward<<<NB, 512, shmem, stream>>>(obs, start, trans, emT, vhist);

  viterbi_backtrace<<<NB, 32, 0, stream>>>(vhist, trT, (float*)d_out);
}